// Lvl1VQ_43860206026821
// MI455X (gfx1250) — compile-verified
//
#include <hip/hip_runtime.h>
#include <hip/hip_bf16.h>
#include <math.h>

typedef __attribute__((ext_vector_type(2))) float v2f;
typedef __attribute__((ext_vector_type(8))) float v8f;
typedef __attribute__((ext_vector_type(4))) unsigned int u32x4;
typedef __attribute__((ext_vector_type(4))) int i32x4;
typedef __attribute__((ext_vector_type(8))) int i32x8;

#define B_ 8
#define D_ 64
#define T_ 2048
#define K_ 8192
#define N_ (B_ * T_)             // 16384 query rows
#define TILE_ELEMS (16 * D_)     // 1024 floats = 4KB per 16-code tile
#define SUBT_ 4                  // k-tiles per staged chunk
#define STAGE_ELEMS (SUBT_ * TILE_ELEMS)   // 4096 floats = 16KB per stage
#define NSTAGE_ (K_ / (16 * SUBT_))        // 128 stages

#if __has_builtin(__builtin_amdgcn_tensor_load_to_lds)
#define HAS_TDM 1
#else
#define HAS_TDM 0
#endif

// d_out layout (floats), reference return order:
//   z_q  : [0, N_*D_)
//   idx  : [N_*D_, N_*D_ + N_)
//   emb  : [N_*D_ + N_, 2*N_*D_ + N_)
//   align: 2*N_*D_ + N_ ; commit: +1

// ---------------------------------------------------------------------------
// Kernel 1: codebook squared norms into workspace; zero the loss accumulators.
// ---------------------------------------------------------------------------
__global__ void __launch_bounds__(256) vq_cnorm_kernel(
    const float* __restrict__ cb, float* __restrict__ cnorm,
    float* __restrict__ out)
{
    int k = blockIdx.x * 256 + threadIdx.x;
    const float* row = cb + (size_t)k * D_;
    float s = 0.f;
#pragma unroll 8
    for (int d = 0; d < D_; ++d) {
        float v = row[d];
        s = fmaf(v, v, s);
    }
    cnorm[k] = s;
    if (blockIdx.x == 0 && threadIdx.x < 2) {
        out[(size_t)2 * N_ * D_ + N_ + threadIdx.x] = 0.f;  // loss slots
    }
}

// ---------------------------------------------------------------------------
// TDM: 1-D DMA of one 64-code x 64-dim fp32 chunk (16KB) into LDS.
// D# per CDNA5 ISA §8 (group0 128b, group1 256b, groups 2/3 zero => <=2D).
// 6-arg builtin form (clang-23 / therock-10.0 toolchain).
// ---------------------------------------------------------------------------
__device__ __forceinline__ void stage_chunk(const float* __restrict__ gsrc,
                                            float* __restrict__ ldst)
{
#if HAS_TDM
    const unsigned long long ga = (unsigned long long)(uintptr_t)gsrc;
    const unsigned int lds_off = (unsigned int)(uintptr_t)ldst;  // low 32 = LDS byte offset
    u32x4 g0;
    g0[0] = 1u;                                   // count=1, is_restore=0, no gather
    g0[1] = lds_off;                              // lds_addr (bytes)
    g0[2] = (unsigned int)ga;                     // global_addr[31:0]
    g0[3] = (unsigned int)((ga >> 32) & 0x01FFFFFFu) | 0x80000000u; // ga[56:32] | type=2
    i32x8 g1;
    g1[0] = (int)(2u << 16);                      // data_size = 4 bytes
    g1[1] = 0;                                    // tensor_dim0[15:0] (=0 for 524288)
    g1[2] = (int)((unsigned)(K_ * D_) >> 16);     // tensor_dim0[31:16]
    g1[3] = (int)((unsigned)STAGE_ELEMS << 16);   // tile_dim0 = 4096 (1-D chunk)
    g1[4] = 0;                                    // tile_dim1 = 0 (1-D), tile_dim2 = 0
    g1[5] = (int)(K_ * D_);                       // tensor_dim0_stride[31:0]
    g1[6] = 0;
    g1[7] = 0;
    i32x4 z4 = {0, 0, 0, 0};                      // groups 2/3 unused (<=2D tensor)
    i32x8 z8 = {0, 0, 0, 0, 0, 0, 0, 0};          // extra group (unused)
    __builtin_amdgcn_tensor_load_to_lds(g0, g1, z4, z4, z8, 0);
#else
    // Fallback: cooperative copy by the issuing wave (32 lanes x b128).
    const int lane = threadIdx.x & 31;
#pragma unroll
    for (int q = 0; q < STAGE_ELEMS / 4; q += 32) {
        ((float4*)ldst)[q + lane] = ((const float4*)gsrc)[q + lane];
    }
#endif
}

__device__ __forceinline__ void wait_chunks_le(int n)
{
#if HAS_TDM
    if (n == 0) __builtin_amdgcn_s_wait_tensorcnt(0);
    else        __builtin_amdgcn_s_wait_tensorcnt(1);
#else
    (void)n;   // plain stores: __syncthreads() ordering suffices
#endif
}

// ---------------------------------------------------------------------------
// Kernel 2: fp32 WMMA distance GEMM + argmin + gather + loss.
// 8 waves/block; each wave owns one 16-row query tile; the block shares
// TDM-staged 64-code chunks through LDS (double buffered, 32KB).
// ---------------------------------------------------------------------------
__global__ void __launch_bounds__(256) vq_main_kernel(
    const float* __restrict__ z_e, const float* __restrict__ cb,
    const float* __restrict__ cnorm, float* __restrict__ out)
{
    __shared__ float sB[2][STAGE_ELEMS];   // 32KB double buffer

    float* z_q   = out;
    float* idxO  = out + (size_t)N_ * D_;
    float* embO  = out + (size_t)N_ * D_ + N_;
    float* lossA = out + (size_t)2 * N_ * D_ + N_;
    float* lossC = lossA + 1;

    const int lane = threadIdx.x & 31;
    const int wave = threadIdx.x >> 5;
    const int tile = blockIdx.x * 8 + wave;   // 16-row tile id (1024 tiles)

    const int m = lane & 15;                  // A: row-in-tile; B/C: col-in-tile
    const int h = lane >> 4;                  // half: selects K-pair / row-half

    // ---- A fragments: a[j] covers dims [4j,4j+3]; lane holds dims 4j+2h, 4j+2h+1
    const int n0 = tile * 16 + m;
    const int b0 = n0 / T_, t0 = n0 % T_;
    const float* zrow = z_e + (size_t)b0 * D_ * T_ + t0;  // d-stride = T_
    v2f a[16];
#pragma unroll
    for (int j = 0; j < 16; ++j) {
        const int d0 = 4 * j + 2 * h;
        a[j].x = zrow[(size_t)d0 * T_];
        a[j].y = zrow[(size_t)(d0 + 1) * T_];
    }

    float runMin[8];
    int   runIdx[8];
#pragma unroll
    for (int r = 0; r < 8; ++r) { runMin[r] = 3.402823466e38f; runIdx[r] = 0; }

    // ---- prologue: stage chunk 0
    if (wave == 0) stage_chunk(cb, &sB[0][0]);

    // ---- sweep 128 chunks of 64 codes, double buffered through LDS
    for (int st = 0; st < NSTAGE_; ++st) {
        const int cur = st & 1;
        if (wave == 0) {
            if (st + 1 < NSTAGE_) {
                stage_chunk(cb + (size_t)(st + 1) * STAGE_ELEMS, &sB[cur ^ 1][0]);
                wait_chunks_le(1);   // chunk st complete (st+1 still in flight)
            } else {
                wait_chunks_le(0);
            }
        }
        __syncthreads();            // sB[cur] visible to all waves

        // hoist the 4 per-sub-tile code norms (clause together, hide latency)
        float cn4[SUBT_];
#pragma unroll
        for (int sub = 0; sub < SUBT_; ++sub)
            cn4[sub] = cnorm[(st * SUBT_ + sub) * 16 + m];

#pragma unroll
        for (int sub = 0; sub < SUBT_; ++sub) {
            const int kt = st * SUBT_ + sub;
            const float* brow = &sB[cur][sub * TILE_ELEMS + m * D_ + 2 * h]; // column = m
            v2f bf[16];
#pragma unroll
            for (int j = 0; j < 16; ++j)
                bf[j] = *(const v2f*)(brow + 4 * j);

            v8f acc = {};
#pragma unroll
            for (int j = 0; j < 16; ++j)
                acc = __builtin_amdgcn_wmma_f32_16x16x4_f32(
                    false, a[j], false, bf[j], (short)0, acc, false, false);

            // score = ||c||^2 - 2 z.c  (z-norm constant per row; drop for argmin)
            const float cn = cn4[sub];
#pragma unroll
            for (int r = 0; r < 8; ++r) {
                const float score = fmaf(-2.0f, acc[r], cn);
                if (score < runMin[r]) { runMin[r] = score; runIdx[r] = kt * 16 + m; }
            }
        }
        __syncthreads();            // all waves done with sB[cur] before reuse
    }

    // ---- per-row reduction across the 16-lane half-groups, then outputs
#pragma unroll
    for (int r = 0; r < 8; ++r) {
        float v = runMin[r];
        int   i = runIdx[r];
#pragma unroll
        for (int mask = 1; mask <= 8; mask <<= 1) {
            const float ov = __shfl_xor(v, mask, 32);
            const int   oi = __shfl_xor(i, mask, 32);
            if (ov < v || (ov == v && oi < i)) { v = ov; i = oi; }
        }
        // all 16 lanes in this half now agree on argmin for row M = r + 8h
        const int nRow = tile * 16 + (r + 8 * h);
        const int bR = nRow / T_, tR = nRow % T_;
        const size_t base = (size_t)bR * D_ * T_ + tR;

        if (m == 0) idxO[nRow] = (float)i;

        // cooperative gather of winning code row + loss term
        const float* cbr = cb + (size_t)i * D_;
        float s4 = 0.f;
#pragma unroll
        for (int q = 0; q < 4; ++q) {
            const int d = m + 16 * q;
            const float cv = cbr[d];
            const size_t o = base + (size_t)d * T_;
            z_q[o]  = cv;    // z_q == emb numerically (straight-through)
            embO[o] = cv;
            const float dz = cv - z_e[o];
            const float d2 = dz * dz;
            s4 = fmaf(d2, d2, s4);
        }
#pragma unroll
        for (int mask = 1; mask <= 8; mask <<= 1)
            s4 += __shfl_xor(s4, mask, 32);
        if (m == 0) {
            const float contrib = sqrtf(s4) * (1.0f / (float)N_);
            atomicAdd(lossA, contrib);   // alignment == commitment numerically
            atomicAdd(lossC, contrib);
        }
    }
}

// ---------------------------------------------------------------------------
extern "C" void kernel_launch(void* const* d_in, const int* in_sizes, int n_in,
                              void* d_out, int out_size, void* d_ws, size_t ws_size,
                              hipStream_t stream)
{
    const float* z_e = (const float*)d_in[0];   // [B, D, T] fp32
    const float* cb  = (const float*)d_in[1];   // [K, D]   fp32
    float* out   = (float*)d_out;
    float* cnorm = (float*)d_ws;                // 8192 floats

    vq_cnorm_kernel<<<K_ / 256, 256, 0, stream>>>(cb, cnorm, out);
    vq_main_kernel<<<(N_ / 16) / 8, 256, 0, stream>>>(z_e, cb, cnorm, out);
}